// XConvolution_14757507629418
// MI455X (gfx1250) — compile-verified
//
#include <hip/hip_runtime.h>
#include <hip/hip_bf16.h>

#define BB   2
#define NP   4096
#define CIN  64
#define KNN  4
#define LP   1024

typedef __attribute__((ext_vector_type(2))) float v2f;
typedef __attribute__((ext_vector_type(8))) float v8f;

__device__ __forceinline__ float eluf(float x) { return x > 0.f ? x : (expf(x) - 1.f); }
__device__ __forceinline__ float bnf(float x, float g, float b, float m, float v) {
  return g * (x - m) * rsqrtf(v + 1e-5f) + b;
}

// ---------------------------------------------------------------------------
// f32 WMMA GEMM, compile-time specialized. One wave computes a 16(M) x 32(N)
// tile with two v8f accumulators sharing each A fragment. K unrolled by 4
// WMMA steps (16 K-values) per iteration so loads cluster and overlap.
//   BT=0: Bm is W[N,K] row-major (C = A*W^T);  BT=1: Bm is [K,N] row-major.
//   GATHER=1: A row index comes from rowIdx[m].
// ---------------------------------------------------------------------------
template <int BT, int BPM, int RELU, int GATHER>
__global__ __launch_bounds__(32) void gemm_wmma(
    const float* __restrict__ A, int lda, const int* __restrict__ rowIdx,
    const float* __restrict__ Bm, int ldb,
    const float* __restrict__ bias,
    float* __restrict__ C, int ldc,
    int M, int N, int K)
{
  const int ntiles = N >> 5;                 // 32-wide N tiles
  const int tile = blockIdx.x;
  const int m0 = (tile / ntiles) << 4;
  const int n0 = (tile % ntiles) << 5;
  const int lane = threadIdx.x;              // wave32
  const int half = lane >> 4;                // lanes 16-31: K = {2,3}
  const int l15  = lane & 15;

  int arow = m0 + l15;                       // A lanes: M = l15, both halves
  if (GATHER) arow = rowIdx[arow];
  const float* __restrict__ Ar = A + (size_t)arow * lda;

  // B pointers for the two 16-wide N subtiles (BT=0 path)
  const float* __restrict__ Bw0 = Bm + (size_t)(n0 + l15) * ldb;
  const float* __restrict__ Bw1 = Bm + (size_t)(n0 + 16 + l15) * ldb;

  v8f acc0 = {};
  v8f acc1 = {};

  for (int k0 = 0; k0 < K; k0 += 16) {
    v2f a[4], b0[4], b1[4];
    #pragma unroll
    for (int u = 0; u < 4; ++u) {
      const int ka = k0 + 4 * u + 2 * half;
      a[u] = *(const v2f*)(Ar + ka);
      if (BT) {                              // Bm is [K,N] row-major
        const float* __restrict__ Bp = Bm + (size_t)ka * ldb + n0 + l15;
        b0[u].x = Bp[0];       b0[u].y = Bp[ldb];
        b1[u].x = Bp[16];      b1[u].y = Bp[ldb + 16];
      } else {                               // Bm is W[N,K] row-major
        b0[u] = *(const v2f*)(Bw0 + ka);
        b1[u] = *(const v2f*)(Bw1 + ka);
      }
    }
    __builtin_prefetch(Ar + k0 + 16, 0, 3);
    #pragma unroll
    for (int u = 0; u < 4; ++u) {
      acc0 = __builtin_amdgcn_wmma_f32_16x16x4_f32(false, a[u], false, b0[u],
                                                   (short)0, acc0, false, false);
      acc1 = __builtin_amdgcn_wmma_f32_16x16x4_f32(false, a[u], false, b1[u],
                                                   (short)0, acc1, false, false);
    }
  }

  const int nA = n0 + l15;
  const int nB = n0 + 16 + l15;
  const float bnA = BPM ? 0.f : bias[nA];
  const float bnB = BPM ? 0.f : bias[nB];
  #pragma unroll
  for (int r = 0; r < 8; ++r) {
    const int m = m0 + r + 8 * half;         // D: lanes>=16 hold M=8..15
    const float bm = BPM ? bias[m] : 0.f;
    float v0 = acc0[r] + (BPM ? bm : bnA);
    float v1 = acc1[r] + (BPM ? bm : bnB);
    if (RELU) { v0 = fmaxf(v0, 0.f); v1 = fmaxf(v1, 0.f); }
    C[(size_t)m * ldc + nA] = v0;
    C[(size_t)m * ldc + nB] = v1;
  }
}

// ---------------------------------------------------------------------------
// Brute-force kNN (K=4, includes self): one thread per query, LDS point tiles
// ---------------------------------------------------------------------------
__global__ __launch_bounds__(256) void knn_kernel(
    const float* __restrict__ pos, const float* __restrict__ qry,
    int N, int Q, int* __restrict__ nbr)
{
  __shared__ float tx[256], ty[256], tz[256];
  const int b  = blockIdx.y;
  const int qi = blockIdx.x * 256 + threadIdx.x;
  const float* __restrict__ pb = pos + (size_t)b * N * 3;
  const float* __restrict__ qb = qry + (size_t)b * Q * 3;
  float qx = 0.f, qy = 0.f, qz = 0.f;
  const bool valid = qi < Q;
  if (valid) { qx = qb[qi*3]; qy = qb[qi*3+1]; qz = qb[qi*3+2]; }

  float d0 = 1e30f, d1 = 1e30f, d2 = 1e30f, d3 = 1e30f;
  int   i0 = 0, i1 = 0, i2 = 0, i3 = 0;

  for (int base = 0; base < N; base += 256) {
    const int m = base + (int)threadIdx.x;
    if (m < N) { tx[threadIdx.x] = pb[m*3]; ty[threadIdx.x] = pb[m*3+1]; tz[threadIdx.x] = pb[m*3+2]; }
    __syncthreads();
    const int lim = min(256, N - base);
    for (int j = 0; j < lim; ++j) {
      const float dx = qx - tx[j], dy = qy - ty[j], dz = qz - tz[j];
      const float d = dx*dx + dy*dy + dz*dz;
      const int mm = base + j;
      if (d < d3) {                    // strict <: earlier index wins ties
        if (d < d0)      { d3=d2;i3=i2; d2=d1;i2=i1; d1=d0;i1=i0; d0=d;i0=mm; }
        else if (d < d1) { d3=d2;i3=i2; d2=d1;i2=i1; d1=d; i1=mm; }
        else if (d < d2) { d3=d2;i3=i2; d2=d; i2=mm; }
        else             { d3=d; i3=mm; }
      }
    }
    __syncthreads();
  }
  if (valid) {
    int* o = nbr + ((size_t)b * Q + qi) * 4;
    o[0]=i0; o[1]=i1; o[2]=i2; o[3]=i3;
  }
}

// ---------------------------------------------------------------------------
// Per-point XConv front half: mlp1, chi-transform, x*T, depthwise -> [160]
// ---------------------------------------------------------------------------
struct PFParams {
  const float *w1,*b1,*bn1g,*bn1b,*bn1m,*bn1v;
  const float *w2,*b2,*bn2g,*bn2b,*bn2m,*bn2v;
  const float *wl,*bl,*bn3g,*bn3b,*bn3m,*bn3v;
  const float *wc1,*bc1,*bn4g,*bn4b,*bn4m,*bn4v;
  const float *wc2,*bc2,*bn5g,*bn5b,*bn5m,*bn5v;
  const float *wd,*bd;
};

__device__ __forceinline__ void chan_out(float* __restrict__ go, int c,
    float a0, float a1, float a2, float a3,
    const float* T, const float* __restrict__ wd, const float* __restrict__ bd)
{
  float xt[4];
  #pragma unroll
  for (int j = 0; j < 4; ++j)
    xt[j] = a0*T[j] + a1*T[4+j] + a2*T[8+j] + a3*T[12+j];
  #pragma unroll
  for (int m = 0; m < 2; ++m) {
    float s = bd[c*2+m];
    #pragma unroll
    for (int k = 0; k < 4; ++k) s += xt[k] * wd[(c*2+m)*4 + k];
    go[c*2+m] = s;
  }
}

__global__ __launch_bounds__(128) void pointfeat_kernel(
    const float* __restrict__ x, const float* __restrict__ pos,
    const int* __restrict__ nbr, PFParams p, float* __restrict__ gfeat)
{
  const int gid = blockIdx.x * 128 + threadIdx.x;
  if (gid >= BB * NP) return;
  const int b = gid >> 12;

  const int* nb = nbr + (size_t)gid * 4;
  int nbk[4] = { nb[0], nb[1], nb[2], nb[3] };
  const float px = pos[gid*3], py = pos[gid*3+1], pz = pos[gid*3+2];

  float rel[4][3];
  #pragma unroll
  for (int k = 0; k < 4; ++k) {
    const float* pm = pos + ((size_t)(b*NP + nbk[k])) * 3;
    rel[k][0] = pm[0]-px; rel[k][1] = pm[1]-py; rel[k][2] = pm[2]-pz;
  }

  // mlp1: Linear(3,16)+ELU+BN, Linear(16,16)+ELU+BN  -> H[k][16]
  float H[4][16];
  #pragma unroll
  for (int k = 0; k < 4; ++k) {
    float hk[16];
    #pragma unroll
    for (int j = 0; j < 16; ++j) {
      float s = p.b1[j] + rel[k][0]*p.w1[j*3] + rel[k][1]*p.w1[j*3+1] + rel[k][2]*p.w1[j*3+2];
      hk[j] = bnf(eluf(s), p.bn1g[j], p.bn1b[j], p.bn1m[j], p.bn1v[j]);
    }
    #pragma unroll
    for (int j = 0; j < 16; ++j) {
      float s = p.b2[j];
      #pragma unroll
      for (int i = 0; i < 16; ++i) s += hk[i] * p.w2[j*16+i];
      H[k][j] = bnf(eluf(s), p.bn2g[j], p.bn2b[j], p.bn2m[j], p.bn2v[j]);
    }
  }

  // chi transform: relf[12] -> Linear(12,16)+ELU+BN -> grouped convs
  float t0[16];
  #pragma unroll
  for (int j = 0; j < 16; ++j) {
    float s = p.bl[j];
    #pragma unroll
    for (int k = 0; k < 4; ++k)
      #pragma unroll
      for (int d = 0; d < 3; ++d) s += rel[k][d] * p.wl[j*12 + k*3 + d];
    t0[j] = bnf(eluf(s), p.bn3g[j], p.bn3b[j], p.bn3m[j], p.bn3v[j]);
  }
  float t1[16];
  #pragma unroll
  for (int g = 0; g < 4; ++g)
    #pragma unroll
    for (int j = 0; j < 4; ++j) {
      const int f = g*4 + j;
      float s = p.bc1[f];
      #pragma unroll
      for (int k = 0; k < 4; ++k) s += t0[g*4+k] * p.wc1[g*16 + j*4 + k];
      t1[f] = bnf(eluf(s), p.bn4g[f], p.bn4b[f], p.bn4m[f], p.bn4v[f]);
    }
  float T[16];
  #pragma unroll
  for (int g = 0; g < 4; ++g)
    #pragma unroll
    for (int j = 0; j < 4; ++j) {
      const int f = g*4 + j;
      float s = p.bc2[f];
      #pragma unroll
      for (int k = 0; k < 4; ++k) s += t1[g*4+k] * p.wc2[g*16 + j*4 + k];
      T[f] = bnf(s, p.bn5g[f], p.bn5b[f], p.bn5m[f], p.bn5v[f]);  // no ELU
    }

  // x_star (80x4) * T (4x4), depthwise conv -> gfeat[gid][160]
  const float* __restrict__ xb = x + ((size_t)b * NP) * CIN;
  float* __restrict__ go = gfeat + (size_t)gid * 160;
  #pragma unroll
  for (int c = 0; c < 16; ++c)
    chan_out(go, c, H[0][c], H[1][c], H[2][c], H[3][c], T, p.wd, p.bd);
  for (int c = 16; c < 80; ++c) {
    const int cc = c - 16;
    chan_out(go, c,
             xb[(size_t)nbk[0]*CIN + cc], xb[(size_t)nbk[1]*CIN + cc],
             xb[(size_t)nbk[2]*CIN + cc], xb[(size_t)nbk[3]*CIN + cc],
             T, p.wd, p.bd);
  }
}

// ---------------------------------------------------------------------------
// Farthest point sampling: one 1024-thread workgroup per batch.
// dmin lives in LDS; argmax via tree reduction (lowest index wins ties).
// ---------------------------------------------------------------------------
__global__ __launch_bounds__(1024) void fps_kernel(
    const float* __restrict__ pos, int* __restrict__ idx,
    float* __restrict__ qbuf, float* __restrict__ qout)
{
  __shared__ float dmin[NP];
  __shared__ float sval[1024];
  __shared__ int   sidx[1024];
  __shared__ int   s_last;
  const int b = blockIdx.x;
  const int tid = threadIdx.x;
  const float* __restrict__ pb = pos + (size_t)b * NP * 3;

  for (int j = tid; j < NP; j += 1024) dmin[j] = 1e30f;
  if (tid == 0) { idx[b*LP] = 0; s_last = 0; }
  __syncthreads();

  for (int i = 1; i < LP; ++i) {
    const int last = s_last;
    const float lx = pb[last*3], ly = pb[last*3+1], lz = pb[last*3+2];
    float bestv = -1.f; int besti = 0;
    #pragma unroll
    for (int t = 0; t < NP/1024; ++t) {
      const int j = tid + t*1024;
      const float dx = pb[j*3]-lx, dy = pb[j*3+1]-ly, dz = pb[j*3+2]-lz;
      const float d = dx*dx + dy*dy + dz*dz;
      const float dm = fminf(dmin[j], d);
      dmin[j] = dm;
      if (dm > bestv) { bestv = dm; besti = j; }
    }
    sval[tid] = bestv; sidx[tid] = besti;
    __syncthreads();
    for (int s = 512; s > 0; s >>= 1) {
      if (tid < s) {
        const float v2 = sval[tid+s]; const int j2 = sidx[tid+s];
        if (v2 > sval[tid] || (v2 == sval[tid] && j2 < sidx[tid])) { sval[tid]=v2; sidx[tid]=j2; }
      }
      __syncthreads();
    }
    if (tid == 0) { s_last = sidx[0]; idx[b*LP + i] = sidx[0]; }
    __syncthreads();
  }

  for (int l = tid; l < LP; l += 1024) {
    const int id = idx[b*LP + l];
    const float X = pb[id*3], Y = pb[id*3+1], Z = pb[id*3+2];
    float* qo = qbuf + ((size_t)b*LP + l)*3;
    qo[0]=X; qo[1]=Y; qo[2]=Z;
    float* qq = qout + ((size_t)b*LP + l)*3;
    qq[0]=X; qq[1]=Y; qq[2]=Z;
  }
}

__global__ void rowidx_kernel(const int* __restrict__ nbr2, int* __restrict__ rowIdx) {
  const int m = blockIdx.x * 256 + threadIdx.x;
  if (m < BB*LP*KNN) rowIdx[m] = (m >> 12) * NP + nbr2[m];
}

// ---------------------------------------------------------------------------
extern "C" void kernel_launch(void* const* d_in, const int* in_sizes, int n_in,
                              void* d_out, int out_size, void* d_ws, size_t ws_size,
                              hipStream_t stream) {
  const float* x   = (const float*)d_in[0];
  const float* pos = (const float*)d_in[1];
  // d_in[2] = batch (layout only; unused)
  PFParams P;
  P.w1  =(const float*)d_in[3];  P.b1  =(const float*)d_in[4];
  P.bn1g=(const float*)d_in[5];  P.bn1b=(const float*)d_in[6];  P.bn1m=(const float*)d_in[7];  P.bn1v=(const float*)d_in[8];
  P.w2  =(const float*)d_in[9];  P.b2  =(const float*)d_in[10];
  P.bn2g=(const float*)d_in[11]; P.bn2b=(const float*)d_in[12]; P.bn2m=(const float*)d_in[13]; P.bn2v=(const float*)d_in[14];
  P.wl  =(const float*)d_in[15]; P.bl  =(const float*)d_in[16];
  P.bn3g=(const float*)d_in[17]; P.bn3b=(const float*)d_in[18]; P.bn3m=(const float*)d_in[19]; P.bn3v=(const float*)d_in[20];
  P.wc1 =(const float*)d_in[21]; P.bc1 =(const float*)d_in[22];
  P.bn4g=(const float*)d_in[23]; P.bn4b=(const float*)d_in[24]; P.bn4m=(const float*)d_in[25]; P.bn4v=(const float*)d_in[26];
  P.wc2 =(const float*)d_in[27]; P.bc2 =(const float*)d_in[28];
  P.bn5g=(const float*)d_in[29]; P.bn5b=(const float*)d_in[30]; P.bn5m=(const float*)d_in[31]; P.bn5v=(const float*)d_in[32];
  P.wd  =(const float*)d_in[33]; P.bd  =(const float*)d_in[34];
  const float* wfc  =(const float*)d_in[35]; const float* bfc  =(const float*)d_in[36];
  const float* fc1w =(const float*)d_in[37]; const float* fc1b =(const float*)d_in[38];
  const float* fc2w =(const float*)d_in[39]; const float* fc2b =(const float*)d_in[40];
  const float* cw   =(const float*)d_in[41]; const float* cb   =(const float*)d_in[42];

  char* w = (char*)d_ws;
  auto alloc = [&](size_t bytes) { char* pp = w; w += (bytes + 255) & ~(size_t)255; return pp; };
  int*   nbr    = (int*)  alloc((size_t)BB*NP*KNN*sizeof(int));
  int*   nbr2   = (int*)  alloc((size_t)BB*LP*KNN*sizeof(int));
  int*   idx    = (int*)  alloc((size_t)BB*LP*sizeof(int));
  int*   rowIdx = (int*)  alloc((size_t)BB*LP*KNN*sizeof(int));
  float* gfeat  = (float*)alloc((size_t)BB*NP*160*sizeof(float));
  float* g      = (float*)alloc((size_t)BB*NP*128*sizeof(float));
  float* qbuf   = (float*)alloc((size_t)BB*LP*3*sizeof(float));
  float* h2a    = (float*)alloc((size_t)BB*LP*KNN*256*sizeof(float));
  float* h2     = (float*)alloc((size_t)BB*LP*KNN*128*sizeof(float));

  float* out = (float*)d_out;                 // [B, L, 128]
  float* qout = out + (size_t)BB*LP*128;      // then q: [B*L, 3]

  // 1. self kNN (K=4, loop=true)
  knn_kernel<<<dim3(NP/256, BB), 256, 0, stream>>>(pos, pos, NP, NP, nbr);
  // 2. per-point XConv front half -> gfeat [8192,160]
  pointfeat_kernel<<<(BB*NP)/128, 128, 0, stream>>>(x, pos, nbr, P, gfeat);
  // 3. g = gfeat @ wfc^T + bfc  -> [8192,128]   (WMMA)
  gemm_wmma<0,0,0,0><<<(BB*NP/16)*(128/32), 32, 0, stream>>>(
      gfeat, 160, nullptr, wfc, 160, bfc, g, 128, BB*NP, 128, 160);
  // 4. FPS -> idx, q positions (also writes q into d_out tail)
  fps_kernel<<<BB, 1024, 0, stream>>>(pos, idx, qbuf, qout);
  // 5. kNN of sampled q against pos
  knn_kernel<<<dim3(LP/256, BB), 256, 0, stream>>>(pos, qbuf, NP, LP, nbr2);
  // 6. gather row indices for f = g[b, nbr2]
  rowidx_kernel<<<(BB*LP*KNN)/256, 256, 0, stream>>>(nbr2, rowIdx);
  // 7. h2a = relu(f @ fc1w^T + fc1b)  [8192,256]   (WMMA, fused gather)
  gemm_wmma<0,0,1,1><<<(BB*LP*KNN/16)*(256/32), 32, 0, stream>>>(
      g, 128, rowIdx, fc1w, 128, fc1b, h2a, 256, BB*LP*KNN, 256, 128);
  // 8. h2 = h2a @ fc2w^T + fc2b  [8192,128]   (WMMA)
  gemm_wmma<0,0,0,0><<<(BB*LP*KNN/16)*(128/32), 32, 0, stream>>>(
      h2a, 256, nullptr, fc2w, 256, fc2b, h2, 128, BB*LP*KNN, 128, 256);
  // 9. out[b] = cw [1024,4096] @ h2[b] [4096,128] + cb  (WMMA, B is [K,N])
  for (int b = 0; b < BB; ++b) {
    gemm_wmma<1,1,0,0><<<(LP/16)*(128/32), 32, 0, stream>>>(
        cw, LP*KNN, nullptr, h2 + (size_t)b*LP*KNN*128, 128,
        cb, out + (size_t)b*LP*128, 128, LP, 128, LP*KNN);
  }
}